// LACEBlock_54271206752699
// MI455X (gfx1250) — compile-verified
//
#include <hip/hip_runtime.h>
#include <hip/hip_bf16.h>

// ---------------- problem constants ----------------
#define BATCH 2
#define DIMC  96
#define DINC  192
#define NSTC  16
#define DTRC  6
#define KDIR  4
#define HH    56
#define WW    56
#define LL    3136      // 56*56
#define NPIX  6272      // BATCH*LL
#define CPROJ 38        // DTR + 2*NST
#define MIDC  24
#define EPSV  1e-5f

typedef __attribute__((ext_vector_type(2))) float v2f;
typedef __attribute__((ext_vector_type(8))) float v8f;

// fp32 WMMA 16x16x4: exact fp32 numerics on the CDNA5 matrix pipe.
__device__ __forceinline__ v8f wmma_f32_k4(v2f a, v2f b, v8f c) {
  return __builtin_amdgcn_wmma_f32_16x16x4_f32(false, a, false, b, (short)0, c,
                                               false, false);
}

__device__ __forceinline__ float wave_red(float v) {
  v += __shfl_xor(v, 16, 32);
  v += __shfl_xor(v, 8, 32);
  v += __shfl_xor(v, 4, 32);
  v += __shfl_xor(v, 2, 32);
  v += __shfl_xor(v, 1, 32);
  return v;
}

__device__ __forceinline__ float half_red(float v) {  // reduce within 16 lanes
  v += __shfl_xor(v, 8, 32);
  v += __shfl_xor(v, 4, 32);
  v += __shfl_xor(v, 2, 32);
  v += __shfl_xor(v, 1, 32);
  return v;
}

__device__ __forceinline__ float silu(float x) { return x / (1.f + __expf(-x)); }

// scan direction l -> pixel index
__device__ __forceinline__ int scan_pix(int kdir, int l) {
  switch (kdir & 3) {
    case 0: return l;
    case 1: { int h = l % WW, w = l / WW; return h * WW + w; }
    case 2: return LL - 1 - l;
    default: { int j = LL - 1 - l; int h = j % WW, w = j / WW; return h * WW + w; }
  }
}

// ---------------- LayerNorm kernels ----------------
// norm1 on NCHW input, one wave per pixel, 3 channels/lane -> (NPIX,96) row-major
__global__ void k_ln_nchw(const float* __restrict__ x, const float* __restrict__ g,
                          const float* __restrict__ bta, float* __restrict__ out) {
  const int lane = threadIdx.x & 31, wave = threadIdx.x >> 5;
  const int p = blockIdx.x * (blockDim.x >> 5) + wave;
  if (p >= NPIX) return;
  const int b = p / LL, q = p % LL;
  const float* xb = x + (size_t)b * DIMC * LL + q;
  float v0 = xb[(size_t)lane * LL];
  float v1 = xb[(size_t)(lane + 32) * LL];
  float v2 = xb[(size_t)(lane + 64) * LL];
  float sum = wave_red(v0 + v1 + v2);
  float sq  = wave_red(v0 * v0 + v1 * v1 + v2 * v2);
  float mean = sum / (float)DIMC;
  float var  = sq / (float)DIMC - mean * mean;
  float rstd = rsqrtf(var + EPSV);
  float* o = out + (size_t)p * DIMC;
  o[lane]      = (v0 - mean) * rstd * g[lane]      + bta[lane];
  o[lane + 32] = (v1 - mean) * rstd * g[lane + 32] + bta[lane + 32];
  o[lane + 64] = (v2 - mean) * rstd * g[lane + 64] + bta[lane + 64];
}

// norm2 on row-major (NPIX,96)
__global__ void k_ln_rows(const float* __restrict__ in, const float* __restrict__ g,
                          const float* __restrict__ bta, float* __restrict__ out) {
  const int lane = threadIdx.x & 31, wave = threadIdx.x >> 5;
  const int p = blockIdx.x * (blockDim.x >> 5) + wave;
  if (p >= NPIX) return;
  const float* r = in + (size_t)p * DIMC;
  float v0 = r[lane], v1 = r[lane + 32], v2 = r[lane + 64];
  float sum = wave_red(v0 + v1 + v2);
  float sq  = wave_red(v0 * v0 + v1 * v1 + v2 * v2);
  float mean = sum / (float)DIMC;
  float var  = sq / (float)DIMC - mean * mean;
  float rstd = rsqrtf(var + EPSV);
  float* o = out + (size_t)p * DIMC;
  o[lane]      = (v0 - mean) * rstd * g[lane]      + bta[lane];
  o[lane + 32] = (v1 - mean) * rstd * g[lane + 32] + bta[lane + 32];
  o[lane + 64] = (v2 - mean) * rstd * g[lane + 64] + bta[lane + 64];
}

// ---------------- in_proj: (NPIX,96)@(96,384) via WMMA f32 ----------------
__global__ void k_in_proj(const float* __restrict__ xn, const float* __restrict__ w,
                          float* __restrict__ xz) {
  const int lane = threadIdx.x & 31, wave = threadIdx.x >> 5;
  const int tile = blockIdx.x * (blockDim.x >> 5) + wave;
  const int NT = (2 * DINC) / 16;  // 24
  if (tile >= (NPIX / 16) * NT) return;
  const int tm = (tile / NT) * 16, tn = (tile % NT) * 16;
  const int lm = lane & 15;
  const int koff = (lane >> 4) << 1;  // 0 or 2
  v8f acc = {};
  for (int k0 = 0; k0 < DIMC; k0 += 4) {
    const int ka = k0 + koff;
    v2f a, b;
    const float* ar = xn + (size_t)(tm + lm) * DIMC + ka;
    a.x = ar[0]; a.y = ar[1];
    const float* br = w + (size_t)(tn + lm) * DIMC + ka;  // B[k][n] = w[n*96+k]
    b.x = br[0]; b.y = br[1];
    acc = wmma_f32_k4(a, b, acc);
  }
  const int rbase = tm + ((lane >> 4) << 3);
  const int col = tn + lm;
#pragma unroll
  for (int r = 0; r < 8; ++r)
    xz[(size_t)(rbase + r) * (2 * DINC) + col] = acc[r];
}

// ---------------- depthwise 3x3 conv + bias + SiLU -> xc (B,192,L) ----------------
__global__ void k_dwconv(const float* __restrict__ xz, const float* __restrict__ cw,
                         const float* __restrict__ cb, float* __restrict__ xc) {
  const int idx = blockIdx.x * blockDim.x + threadIdx.x;
  if (idx >= BATCH * DINC * LL) return;
  const int q = idx % LL;
  const int d = (idx / LL) % DINC;
  const int b = idx / (LL * DINC);
  const int h = q / WW, w0 = q % WW;
  float s = cb[d];
#pragma unroll
  for (int dy = -1; dy <= 1; ++dy) {
    const int hh = h + dy;
    if (hh < 0 || hh >= HH) continue;
#pragma unroll
    for (int dx = -1; dx <= 1; ++dx) {
      const int ww_ = w0 + dx;
      if (ww_ < 0 || ww_ >= WW) continue;
      s += xz[((size_t)(b * LL + hh * WW + ww_)) * (2 * DINC) + d] *
           cw[d * 9 + (dy + 1) * 3 + (dx + 1)];
    }
  }
  xc[idx] = silu(s);
}

// ---------------- x_proj: per (b,dir) (L,192)@(192,38) via WMMA ----------------
__global__ void k_xproj(const float* __restrict__ xc, const float* __restrict__ xpw,
                        float* __restrict__ xdbl) {
  const int lane = threadIdx.x & 31, wave = threadIdx.x >> 5;
  int tile = blockIdx.x * (blockDim.x >> 5) + wave;
  const int NT = 3, MT = LL / 16;  // N padded 38 -> 48
  if (tile >= BATCH * KDIR * MT * NT) return;
  const int nt = tile % NT; tile /= NT;
  const int mt = tile % MT; tile /= MT;
  const int kdir = tile % KDIR;
  const int b = tile / KDIR;
  const int tm = mt * 16, tn = nt * 16;
  const int lm = lane & 15;
  const int koff = (lane >> 4) << 1;
  const int lrow = tm + lm;
  const int p = scan_pix(kdir, lrow);
  const int col = tn + lm;
  const float* xcb = xc + (size_t)b * DINC * LL;
  v8f acc = {};
  for (int k0 = 0; k0 < DINC; k0 += 4) {
    const int ka = k0 + koff;
    v2f a, bb;
    a.x = xcb[(size_t)ka * LL + p];
    a.y = xcb[(size_t)(ka + 1) * LL + p];
    const bool ok = (col < CPROJ);
    bb.x = ok ? xpw[((size_t)kdir * CPROJ + col) * DINC + ka] : 0.f;
    bb.y = ok ? xpw[((size_t)kdir * CPROJ + col) * DINC + ka + 1] : 0.f;
    acc = wmma_f32_k4(a, bb, acc);
  }
  if (col < CPROJ) {
    float* ob = xdbl + ((size_t)(b * KDIR + kdir)) * CPROJ * LL + (size_t)col * LL;
    const int rbase = tm + ((lane >> 4) << 3);
#pragma unroll
    for (int r = 0; r < 8; ++r) ob[rbase + r] = acc[r];
  }
}

// ---------------- selective scan: one wave handles 2 channels x 16 states ----------------
__global__ void k_scan(const float* __restrict__ xc, const float* __restrict__ xdbl,
                       const float* __restrict__ dtw, const float* __restrict__ dtb,
                       const float* __restrict__ alog, const float* __restrict__ Ds,
                       float* __restrict__ ys) {
  const int lane = threadIdx.x & 31, wave = threadIdx.x >> 5;
  int wg = blockIdx.x * (blockDim.x >> 5) + wave;
  if (wg >= BATCH * KDIR * (DINC / 2)) return;
  const int pairid = wg % (DINC / 2); wg /= (DINC / 2);
  const int kdir = wg % KDIR;
  const int b = wg / KDIR;
  const int n = lane & 15;
  const int d = pairid * 2 + (lane >> 4);

  const int kd = kdir * DINC + d;
  const float Aval = -__expf(alog[(size_t)kd * NSTC + n]);
  const float Dval = Ds[kd];
  const float bias = dtb[kd];
  float dw[DTRC];
#pragma unroll
  for (int r = 0; r < DTRC; ++r) dw[r] = dtw[(size_t)kd * DTRC + r];

  const float* xrow  = xc + ((size_t)(b * DINC + d)) * LL;
  const float* dbase = xdbl + ((size_t)(b * KDIR + kdir)) * CPROJ * LL;
  const float* Brow  = dbase + (size_t)(DTRC + n) * LL;
  const float* Crow  = dbase + (size_t)(DTRC + NSTC + n) * LL;
  float* yrow = ys + ((size_t)((b * KDIR + kdir) * DINC + d)) * LL;

  float hstate = 0.f;
  for (int l = 0; l < LL; ++l) {
    if (((l & 63) == 0) && (l + 64 < LL)) {
      __builtin_prefetch(Brow + l + 64, 0, 0);   // -> global_prefetch_b8
      __builtin_prefetch(Crow + l + 64, 0, 0);
    }
    const int p = scan_pix(kdir, l);
    const float x = xrow[p];
    float draw = bias;
#pragma unroll
    for (int r = 0; r < DTRC; ++r) draw += dw[r] * dbase[(size_t)r * LL + l];
    const float delta = (draw > 20.f) ? draw : log1pf(__expf(draw));
    const float dA = __expf(delta * Aval);
    hstate = dA * hstate + delta * Brow[l] * x;
    float contrib = half_red(hstate * Crow[l]);
    if (n == 0) yrow[l] = contrib + Dval * x;
  }
}

// ---------------- gather 4 directions + out_norm LN + SiLU(z) gate ----------------
__global__ void k_merge(const float* __restrict__ ys, const float* __restrict__ xz,
                        const float* __restrict__ ong, const float* __restrict__ onb,
                        float* __restrict__ g) {
  const int lane = threadIdx.x & 31, wave = threadIdx.x >> 5;
  const int p = blockIdx.x * (blockDim.x >> 5) + wave;
  if (p >= NPIX) return;
  const int b = p / LL, q = p % LL;
  const int h = q / WW, w0 = q % WW;
  const int l1 = w0 * WW + h;     // inverse of dir-1 map
  const int l2 = LL - 1 - q;      // inverse of dir-2 map
  const int l3 = LL - 1 - l1;     // inverse of dir-3 map
  const float* base = ys + (size_t)b * KDIR * DINC * LL;
  float vals[6];
  float sum = 0.f, sq = 0.f;
#pragma unroll
  for (int j = 0; j < 6; ++j) {
    const int d = lane + j * 32;
    float v = base[(size_t)(0 * DINC + d) * LL + q]
            + base[(size_t)(1 * DINC + d) * LL + l1]
            + base[(size_t)(2 * DINC + d) * LL + l2]
            + base[(size_t)(3 * DINC + d) * LL + l3];
    vals[j] = v; sum += v; sq += v * v;
  }
  sum = wave_red(sum); sq = wave_red(sq);
  const float mean = sum / (float)DINC;
  const float var  = sq / (float)DINC - mean * mean;
  const float rstd = rsqrtf(var + EPSV);
#pragma unroll
  for (int j = 0; j < 6; ++j) {
    const int d = lane + j * 32;
    const float zv = xz[(size_t)p * (2 * DINC) + DINC + d];
    g[(size_t)p * DINC + d] =
        ((vals[j] - mean) * rstd * ong[d] + onb[d]) * silu(zv);
  }
}

// ---------------- out_proj GEMM + scaled residual ----------------
__global__ void k_outproj(const float* __restrict__ g, const float* __restrict__ w,
                          const float* __restrict__ xin, const float* __restrict__ s1,
                          float* __restrict__ vl) {
  const int lane = threadIdx.x & 31, wave = threadIdx.x >> 5;
  const int tile = blockIdx.x * (blockDim.x >> 5) + wave;
  const int NT = DIMC / 16;  // 6
  if (tile >= (NPIX / 16) * NT) return;
  const int tm = (tile / NT) * 16, tn = (tile % NT) * 16;
  const int lm = lane & 15;
  const int koff = (lane >> 4) << 1;
  v8f acc = {};
  for (int k0 = 0; k0 < DINC; k0 += 4) {
    const int ka = k0 + koff;
    v2f a, b;
    const float* ar = g + (size_t)(tm + lm) * DINC + ka;
    a.x = ar[0]; a.y = ar[1];
    const float* br = w + (size_t)(tn + lm) * DINC + ka;
    b.x = br[0]; b.y = br[1];
    acc = wmma_f32_k4(a, b, acc);
  }
  const int rbase = tm + ((lane >> 4) << 3);
  const int col = tn + lm;
#pragma unroll
  for (int r = 0; r < 8; ++r) {
    const int row = rbase + r;
    const int b_ = row / LL, q = row % LL;
    vl[(size_t)row * DIMC + col] =
        acc[r] + s1[col] * xin[((size_t)(b_ * DIMC + col)) * LL + q];
  }
}

// ---------------- lc1: 1x1 conv 96->24 + BN + ReLU (WMMA, N padded to 32) ----------------
__global__ void k_lc1(const float* __restrict__ vb, const float* __restrict__ w1,
                      const float* __restrict__ bg, const float* __restrict__ bb,
                      const float* __restrict__ bm, const float* __restrict__ bv,
                      float* __restrict__ h1) {
  const int lane = threadIdx.x & 31, wave = threadIdx.x >> 5;
  const int tile = blockIdx.x * (blockDim.x >> 5) + wave;
  const int NT = 2;
  if (tile >= (NPIX / 16) * NT) return;
  const int tm = (tile / NT) * 16, tn = (tile % NT) * 16;
  const int lm = lane & 15;
  const int koff = (lane >> 4) << 1;
  const int col = tn + lm;
  v8f acc = {};
  for (int k0 = 0; k0 < DIMC; k0 += 4) {
    const int ka = k0 + koff;
    v2f a, b;
    const float* ar = vb + (size_t)(tm + lm) * DIMC + ka;
    a.x = ar[0]; a.y = ar[1];
    const bool ok = (col < MIDC);
    b.x = ok ? w1[(size_t)col * DIMC + ka] : 0.f;
    b.y = ok ? w1[(size_t)col * DIMC + ka + 1] : 0.f;
    acc = wmma_f32_k4(a, b, acc);
  }
  if (col < MIDC) {
    const float rstd = rsqrtf(bv[col] + EPSV);
    const int rbase = tm + ((lane >> 4) << 3);
#pragma unroll
    for (int r = 0; r < 8; ++r) {
      float t = (acc[r] - bm[col]) * rstd * bg[col] + bb[col];
      h1[(size_t)(rbase + r) * MIDC + col] = fmaxf(t, 0.f);
    }
  }
}

// ---------------- 3x3 conv 24->24 + BN + ReLU ----------------
__global__ void k_conv3(const float* __restrict__ h1, const float* __restrict__ w2,
                        const float* __restrict__ bg, const float* __restrict__ bb,
                        const float* __restrict__ bm, const float* __restrict__ bv,
                        float* __restrict__ h2) {
  const int idx = blockIdx.x * blockDim.x + threadIdx.x;
  if (idx >= NPIX * MIDC) return;
  const int mo = idx % MIDC;
  const int p = idx / MIDC;
  const int b = p / LL, q = p % LL;
  const int h = q / WW, w0 = q % WW;
  float s = 0.f;
  for (int dy = -1; dy <= 1; ++dy) {
    const int hh = h + dy;
    if (hh < 0 || hh >= HH) continue;
    for (int dx = -1; dx <= 1; ++dx) {
      const int ww_ = w0 + dx;
      if (ww_ < 0 || ww_ >= WW) continue;
      const float* pin = h1 + ((size_t)(b * LL + hh * WW + ww_)) * MIDC;
      const float* pw = w2 + (size_t)mo * MIDC * 9 + (dy + 1) * 3 + (dx + 1);
#pragma unroll
      for (int mi = 0; mi < MIDC; ++mi) s += pin[mi] * pw[mi * 9];
    }
  }
  float t = (s - bm[mo]) * rsqrtf(bv[mo] + EPSV) * bg[mo] + bb[mo];
  h2[(size_t)p * MIDC + mo] = fmaxf(t, 0.f);
}

// ---------------- lc3: 1x1 conv 24->96 + BN (WMMA, K=24) ----------------
__global__ void k_lc3(const float* __restrict__ h2, const float* __restrict__ w3,
                      const float* __restrict__ bg, const float* __restrict__ bb,
                      const float* __restrict__ bm, const float* __restrict__ bv,
                      float* __restrict__ h3) {
  const int lane = threadIdx.x & 31, wave = threadIdx.x >> 5;
  const int tile = blockIdx.x * (blockDim.x >> 5) + wave;
  const int NT = DIMC / 16;
  if (tile >= (NPIX / 16) * NT) return;
  const int tm = (tile / NT) * 16, tn = (tile % NT) * 16;
  const int lm = lane & 15;
  const int koff = (lane >> 4) << 1;
  const int col = tn + lm;
  v8f acc = {};
  for (int k0 = 0; k0 < MIDC; k0 += 4) {
    const int ka = k0 + koff;
    v2f a, b;
    const float* ar = h2 + (size_t)(tm + lm) * MIDC + ka;
    a.x = ar[0]; a.y = ar[1];
    const float* br = w3 + (size_t)col * MIDC + ka;
    b.x = br[0]; b.y = br[1];
    acc = wmma_f32_k4(a, b, acc);
  }
  const float rstd = rsqrtf(bv[col] + EPSV);
  const int rbase = tm + ((lane >> 4) << 3);
#pragma unroll
  for (int r = 0; r < 8; ++r)
    h3[(size_t)(rbase + r) * DIMC + col] =
        (acc[r] - bm[col]) * rstd * bg[col] + bb[col];
}

// ---------------- SE: global average pool ----------------
__global__ void k_sepool(const float* __restrict__ h3, float* __restrict__ pool) {
  const int lane = threadIdx.x & 31, wave = threadIdx.x >> 5;
  const int bc = blockIdx.x * (blockDim.x >> 5) + wave;
  if (bc >= BATCH * DIMC) return;
  const int b = bc / DIMC, c = bc % DIMC;
  float s = 0.f;
  for (int q = lane; q < LL; q += 32) s += h3[((size_t)(b * LL + q)) * DIMC + c];
  s = wave_red(s);
  if (lane == 0) pool[bc] = s / (float)LL;
}

// ---------------- SE MLP (uses LDS) ----------------
__global__ void k_semlp(const float* __restrict__ pool, const float* __restrict__ w1,
                        const float* __restrict__ w2, float* __restrict__ sescale) {
  __shared__ float hid[MIDC];
  const int b = blockIdx.x;
  const int t = threadIdx.x;  // blockDim.x == 96
  if (t < MIDC) {
    float s = 0.f;
    for (int c = 0; c < DIMC; ++c) s += pool[b * DIMC + c] * w1[t * DIMC + c];
    hid[t] = fmaxf(s, 0.f);
  }
  __syncthreads();
  float s = 0.f;
#pragma unroll
  for (int m = 0; m < MIDC; ++m) s += hid[m] * w2[t * MIDC + m];
  sescale[b * DIMC + t] = 1.f / (1.f + __expf(-s));
}

// ---------------- final: h3*se + scale2*vl, NCHW output ----------------
__global__ void k_final(const float* __restrict__ h3, const float* __restrict__ sescale,
                        const float* __restrict__ scale2, const float* __restrict__ vl,
                        float* __restrict__ out) {
  const int idx = blockIdx.x * blockDim.x + threadIdx.x;
  if (idx >= BATCH * DIMC * LL) return;
  const int q = idx % LL;
  const int c = (idx / LL) % DIMC;
  const int b = idx / (LL * DIMC);
  const size_t prow = (size_t)(b * LL + q) * DIMC + c;
  out[idx] = h3[prow] * sescale[b * DIMC + c] + scale2[c] * vl[prow];
}

// ---------------- host launcher ----------------
extern "C" void kernel_launch(void* const* d_in, const int* in_sizes, int n_in,
                              void* d_out, int out_size, void* d_ws, size_t ws_size,
                              hipStream_t stream) {
  (void)in_sizes; (void)n_in; (void)out_size; (void)ws_size;
  const float* x        = (const float*)d_in[0];
  const float* n1g      = (const float*)d_in[1];
  const float* n1b      = (const float*)d_in[2];
  const float* in_proj  = (const float*)d_in[3];
  const float* conv_w   = (const float*)d_in[4];
  const float* conv_b   = (const float*)d_in[5];
  const float* xproj_w  = (const float*)d_in[6];
  const float* dt_w     = (const float*)d_in[7];
  const float* dt_b     = (const float*)d_in[8];
  const float* A_log    = (const float*)d_in[9];
  const float* Ds       = (const float*)d_in[10];
  const float* on_g     = (const float*)d_in[11];
  const float* on_b     = (const float*)d_in[12];
  const float* out_proj = (const float*)d_in[13];
  const float* scale1   = (const float*)d_in[14];
  const float* n2g      = (const float*)d_in[15];
  const float* n2b      = (const float*)d_in[16];
  const float* lc_w1    = (const float*)d_in[17];
  const float* lc_g1    = (const float*)d_in[18];
  const float* lc_b1    = (const float*)d_in[19];
  const float* lc_m1    = (const float*)d_in[20];
  const float* lc_v1    = (const float*)d_in[21];
  const float* lc_w2    = (const float*)d_in[22];
  const float* lc_g2    = (const float*)d_in[23];
  const float* lc_b2    = (const float*)d_in[24];
  const float* lc_m2    = (const float*)d_in[25];
  const float* lc_v2    = (const float*)d_in[26];
  const float* lc_w3    = (const float*)d_in[27];
  const float* lc_g3    = (const float*)d_in[28];
  const float* lc_b3    = (const float*)d_in[29];
  const float* lc_m3    = (const float*)d_in[30];
  const float* lc_v3    = (const float*)d_in[31];
  const float* se_w1    = (const float*)d_in[32];
  const float* se_w2    = (const float*)d_in[33];
  const float* scale2   = (const float*)d_in[34];
  float* out = (float*)d_out;

  float* ws = (float*)d_ws;
  size_t o = 0;
  float* xn   = ws + o; o += (size_t)NPIX * DIMC;             // 602112
  float* xz   = ws + o; o += (size_t)NPIX * 2 * DINC;         // 2408448
  float* xc   = ws + o; o += (size_t)BATCH * DINC * LL;       // 1204224
  float* xdbl = ws + o; o += (size_t)BATCH * KDIR * CPROJ * LL; // 953344
  float* ysb  = ws + o; o += (size_t)BATCH * KDIR * DINC * LL;  // 4816896
  float* gbuf = ws + o; o += (size_t)NPIX * DINC;             // 1204224
  float* vl   = ws + o; o += (size_t)NPIX * DIMC;
  float* vb   = ws + o; o += (size_t)NPIX * DIMC;
  float* h1b  = ws + o; o += (size_t)NPIX * MIDC;
  float* h2b  = ws + o; o += (size_t)NPIX * MIDC;
  float* h3b  = ws + o; o += (size_t)NPIX * DIMC;
  float* poolb = ws + o; o += (size_t)BATCH * DIMC;
  float* seb   = ws + o; o += (size_t)BATCH * DIMC;           // total ~53 MB (< 192 MB L2)

  // 1. norm1
  k_ln_nchw<<<NPIX / 8, 256, 0, stream>>>(x, n1g, n1b, xn);
  // 2. in_proj GEMM (WMMA f32)
  k_in_proj<<<(NPIX / 16) * 24 / 8, 256, 0, stream>>>(xn, in_proj, xz);
  // 3. depthwise conv + silu
  k_dwconv<<<(BATCH * DINC * LL) / 256, 256, 0, stream>>>(xz, conv_w, conv_b, xc);
  // 4. x_proj GEMM (WMMA f32)
  k_xproj<<<(BATCH * KDIR * (LL / 16) * 3) / 8, 256, 0, stream>>>(xc, xproj_w, xdbl);
  // 5. fused selective scan (delta/softplus/exp/scan)
  k_scan<<<(BATCH * KDIR * (DINC / 2)) / 8, 256, 0, stream>>>(xc, xdbl, dt_w, dt_b,
                                                              A_log, Ds, ysb);
  // 6. merge directions + out_norm + gate
  k_merge<<<NPIX / 8, 256, 0, stream>>>(ysb, xz, on_g, on_b, gbuf);
  // 7. out_proj GEMM + scaled residual (WMMA f32)
  k_outproj<<<(NPIX / 16) * 6 / 8, 256, 0, stream>>>(gbuf, out_proj, x, scale1, vl);
  // 8. norm2
  k_ln_rows<<<NPIX / 8, 256, 0, stream>>>(vl, n2g, n2b, vb);
  // 9. 1x1 conv + BN + relu (WMMA f32)
  k_lc1<<<(NPIX / 16) * 2 / 8, 256, 0, stream>>>(vb, lc_w1, lc_g1, lc_b1, lc_m1,
                                                 lc_v1, h1b);
  // 10. 3x3 conv + BN + relu
  k_conv3<<<(NPIX * MIDC) / 256, 256, 0, stream>>>(h1b, lc_w2, lc_g2, lc_b2, lc_m2,
                                                   lc_v2, h2b);
  // 11. 1x1 conv + BN (WMMA f32)
  k_lc3<<<(NPIX / 16) * 6 / 8, 256, 0, stream>>>(h2b, lc_w3, lc_g3, lc_b3, lc_m3,
                                                 lc_v3, h3b);
  // 12. SE pool
  k_sepool<<<(BATCH * DIMC) / 8, 256, 0, stream>>>(h3b, poolb);
  // 13. SE MLP
  k_semlp<<<BATCH, 96, 0, stream>>>(poolb, se_w1, se_w2, seb);
  // 14. final mix, NCHW out
  k_final<<<(BATCH * DIMC * LL) / 256, 256, 0, stream>>>(h3b, seb, scale2, vl, out);
}